// CacheDiffSimWithFrictionBatch_5806795784812
// MI455X (gfx1250) — compile-verified
//
#include <hip/hip_runtime.h>

#define DT_STEP 5e-4f
#define BOUNDW  3
#define BLOCKT  256
#define WAVESZ  32
#define WPB     (BLOCKT / WAVESZ)          // 8 waves per block
#define CHUNK_NODES  128                   // nodes per wave-chunk
#define CHUNK_BYTES  (CHUNK_NODES * 12)    // 1536 B = 3 x b128 per lane
#define CHUNK_F4     (CHUNK_BYTES / 16)    // 96 float4

// --- grid op: floor Coulomb friction (+y normal) then boundary clamp -------
__device__ __forceinline__ void apply_grid_op(
    float& vx, float& vy, float& vz,
    int gi, int gj, int gk, int G,
    float mu, float fl, float dx) {
  float ypos    = (float)gj * dx;
  bool  contact = (ypos <= fl + (float)BOUNDW * dx) && (vy < 0.0f);
  float s       = fmaf(vx, vx, fmaf(vz, vz, 1e-10f));
  float inv_vtn = __builtin_amdgcn_rcpf(__builtin_amdgcn_sqrtf(s));
  float scale   = fmaxf(0.0f, fmaf(mu * vy, inv_vtn, 1.0f));
  float sx = vx * scale, sz = vz * scale;
  vx = contact ? sx  : vx;
  vy = contact ? 0.f : vy;
  vz = contact ? sz  : vz;
  int hi = G - BOUNDW;
  vx = ((gi < BOUNDW && vx < 0.0f) || (gi >= hi && vx > 0.0f)) ? 0.0f : vx;
  vy = ((gj < BOUNDW && vy < 0.0f) || (gj >= hi && vy > 0.0f)) ? 0.0f : vy;
  vz = ((gk < BOUNDW && vz < 0.0f) || (gk >= hi && vz > 0.0f)) ? 0.0f : vz;
}

// Low 32 bits of a generic (flat) pointer into LDS == wave-relative LDS byte
// address (flat LDS aperture keeps the offset in addr[31:0]).
__device__ __forceinline__ unsigned lds_byte_addr(const void* p) {
  return (unsigned)(size_t)p;
}

// Issue one 1536B chunk as 3 async b128 loads per lane (ASYNCcnt += 3).
// pred is read exactly once -> NT hint so it doesn't evict the ws intermediate
// from L2 (which pass 2 re-reads ~3.2x).
__device__ __forceinline__ void issue_chunk_async(const float* __restrict__ src,
                                                  unsigned gByteBase,
                                                  unsigned ldsByteBase,
                                                  int lane) {
#pragma unroll
  for (int k = 0; k < 3; ++k) {
    unsigned off  = (unsigned)(lane * 16 + k * 512);
    unsigned loff = ldsByteBase + off;
    unsigned goff = gByteBase + off;
    asm volatile("global_load_async_to_lds_b128 %0, %1, %2 th:TH_LOAD_NT"
                 :: "v"(loff), "v"(goff), "s"(src)
                 : "memory");
  }
}

// Transform 4 whole nodes (12 floats) per lane out of LDS, store b128 x3.
__device__ __forceinline__ void process_chunk(
    const float4* __restrict__ buf, float* __restrict__ out,
    const float* __restrict__ friction, const float* __restrict__ cfloor,
    int nodeBase, int lane, int gShift, int gMask, float dx) {
  int n0 = nodeBase + lane * 4;
  float4 a  = buf[lane * 3 + 0];
  float4 b4 = buf[lane * 3 + 1];
  float4 c4 = buf[lane * 3 + 2];
  int   bidx = n0 >> (3 * gShift);           // batch uniform across the 4 nodes
  float mu = friction[bidx], fl = cfloor[bidx];
  float v[12] = {a.x, a.y, a.z, a.w, b4.x, b4.y, b4.z, b4.w,
                 c4.x, c4.y, c4.z, c4.w};
#pragma unroll
  for (int t = 0; t < 4; ++t) {
    int n  = n0 + t;
    int gk = n & gMask;
    int gj = (n >> gShift) & gMask;
    int gi = (n >> (2 * gShift)) & gMask;
    apply_grid_op(v[3 * t], v[3 * t + 1], v[3 * t + 2],
                  gi, gj, gk, gMask + 1, mu, fl, dx);
  }
  // ws intermediate: keep RT (must stay L2-resident for the g2p gathers)
  float4* op = (float4*)(out + (size_t)n0 * 3);
  op[0] = make_float4(v[0], v[1], v[2],  v[3]);
  op[1] = make_float4(v[4], v[5], v[6],  v[7]);
  op[2] = make_float4(v[8], v[9], v[10], v[11]);
}

// --- pass 1 (pow2 G): async double-buffered streaming grid transform -------
__global__ __launch_bounds__(BLOCKT)
void grid_op_async_kernel(const float* __restrict__ pred,
                          const float* __restrict__ friction,
                          const float* __restrict__ cfloor,
                          float* __restrict__ out,
                          int totalNodes, int gShift, int gMask, float dx) {
  __shared__ float4 smem[WPB * 2 * CHUNK_F4];           // 24 KB, per-wave buffers
  const int lane   = threadIdx.x & (WAVESZ - 1);
  const int wave   = threadIdx.x >> 5;
  const int gwave  = blockIdx.x * WPB + wave;
  const int nWaves = gridDim.x * WPB;
  const int nChunks = totalNodes / CHUNK_NODES;

  float4* buf0 = &smem[wave * 2 * CHUNK_F4];
  float4* buf1 = buf0 + CHUNK_F4;
  unsigned lds0 = lds_byte_addr(buf0);
  unsigned lds1 = lds_byte_addr(buf1);

  int c = gwave;
  if (c < nChunks)
    issue_chunk_async(pred, (unsigned)c * CHUNK_BYTES, lds0, lane);
  int cur = 0;
  for (; c < nChunks; c += nWaves) {
    int cn = c + nWaves;
    const float4* curBuf = cur ? buf1 : buf0;
    unsigned      nxtLds = cur ? lds0 : lds1;
    if (cn < nChunks) {
      issue_chunk_async(pred, (unsigned)cn * CHUNK_BYTES, nxtLds, lane);
      asm volatile("s_wait_asynccnt 0x3" ::: "memory");  // drain chunk c (in-order)
    } else {
      asm volatile("s_wait_asynccnt 0x0" ::: "memory");
    }
    process_chunk(curBuf, out, friction, cfloor, c * CHUNK_NODES,
                  lane, gShift, gMask, dx);
    cur ^= 1;
  }
  // tail nodes (totalNodes % 128) handled by global wave 0, plain path
  int mainNodes = nChunks * CHUNK_NODES;
  if (gwave == 0) {
    for (int n = mainNodes + lane; n < totalNodes; n += WAVESZ) {
      const float* p = pred + (size_t)n * 3;
      float vx = p[0], vy = p[1], vz = p[2];
      int gk = n & gMask, gj = (n >> gShift) & gMask;
      int gi = (n >> (2 * gShift)) & gMask, bb = n >> (3 * gShift);
      apply_grid_op(vx, vy, vz, gi, gj, gk, gMask + 1, friction[bb], cfloor[bb], dx);
      float* o = out + (size_t)n * 3;
      o[0] = vx; o[1] = vy; o[2] = vz;
    }
  }
}

// --- pass 1 fallback (non-pow2 G): plain streaming kernel ------------------
__global__ __launch_bounds__(BLOCKT)
void grid_op_kernel(const float* __restrict__ pred,
                    const float* __restrict__ friction,
                    const float* __restrict__ cfloor,
                    float* __restrict__ out,
                    int Bn, int G, float dx) {
  int nid   = blockIdx.x * blockDim.x + threadIdx.x;
  int total = Bn * G * G * G;
  if (nid >= total) return;
  const float* p = pred + (size_t)nid * 3;
  float vx = __builtin_nontemporal_load(p + 0);
  float vy = __builtin_nontemporal_load(p + 1);
  float vz = __builtin_nontemporal_load(p + 2);
  int gk = nid % G;
  int t  = nid / G;
  int gj = t % G;  t /= G;
  int gi = t % G;
  int b  = t / G;
  apply_grid_op(vx, vy, vz, gi, gj, gk, G, friction[b], cfloor[b], dx);
  float* o = out + (size_t)nid * 3;
  o[0] = vx; o[1] = vy; o[2] = vz;
}

__device__ __forceinline__ void bspline_w(float f, float w[3]) {
  float a = 1.5f - f, b = f - 1.0f, c = f - 0.5f;
  w[0] = 0.5f * a * a;
  w[1] = 0.75f - b * b;
  w[2] = 0.5f * c * c;
}

// --- pass 2: G2P 3x3x3 gather (PIC). FUSED applies grid-op inline (fallback).
template <bool FUSED>
__global__ __launch_bounds__(BLOCKT)
void g2p_kernel(const float* __restrict__ x,
                const float* __restrict__ grid,
                const float* __restrict__ friction,
                const float* __restrict__ cfloor,
                const int* __restrict__ statics,
                float* __restrict__ x_out,
                float* __restrict__ v_out,
                int Bn, int Np, int G, float dx, float inv_dx) {
  int pid   = blockIdx.x * blockDim.x + threadIdx.x;
  int total = Bn * Np;
  if (pid >= total) return;
  int b = pid / Np;

  const float* xp = x + (size_t)pid * 3;
  __builtin_prefetch(xp + 3 * 256 * 8, 0, 0);   // gfx1250 global_prefetch_b8
  // particle stream: touched once -> NT, keep L2 for the grid
  float x0 = __builtin_nontemporal_load(xp + 0);
  float x1 = __builtin_nontemporal_load(xp + 1);
  float x2 = __builtin_nontemporal_load(xp + 2);
  float px = x0 * inv_dx, py = x1 * inv_dx, pz = x2 * inv_dx;
  int bx = (int)floorf(px - 0.5f);
  int by = (int)floorf(py - 0.5f);
  int bz = (int)floorf(pz - 0.5f);
  float fxx = px - (float)bx, fxy = py - (float)by, fxz = pz - (float)bz;

  float wx[3], wy[3], wz[3];
  bspline_w(fxx, wx); bspline_w(fxy, wy); bspline_w(fxz, wz);

  int ci[3], cj[3], ck[3], rowI[3], rowJ[3], rowK[3];
#pragma unroll
  for (int t = 0; t < 3; ++t) {
    ci[t] = min(max(bx + t, 0), G - 1);
    cj[t] = min(max(by + t, 0), G - 1);
    ck[t] = min(max(bz + t, 0), G - 1);
    rowI[t] = ci[t] * G * G * 3;
    rowJ[t] = cj[t] * G * 3;
    rowK[t] = ck[t] * 3;
  }

  const float* gb = grid + (size_t)b * G * G * G * 3;
  float mu = 0.0f, fl = 0.0f;
  if (FUSED) { mu = friction[b]; fl = cfloor[b]; }

  float ax = 0.0f, ay = 0.0f, az = 0.0f;
#pragma unroll
  for (int i = 0; i < 3; ++i) {
#pragma unroll
    for (int j = 0; j < 3; ++j) {
      float wij = wx[i] * wy[j];
      const float* row = gb + rowI[i] + rowJ[j];
#pragma unroll
      for (int k = 0; k < 3; ++k) {
        const float* g = row + rowK[k];
        float gx = g[0], gy = g[1], gz = g[2];    // b96 gather, RT -> hits L2
        if (FUSED) apply_grid_op(gx, gy, gz, ci[i], cj[j], ck[k], G, mu, fl, dx);
        float w = wij * wz[k];
        ax = fmaf(w, gx, ax);
        ay = fmaf(w, gy, ay);
        az = fmaf(w, gz, az);
      }
    }
  }

  float m = (float)__builtin_nontemporal_load(statics + pid);
  ax *= m; ay *= m; az *= m;
  // outputs are write-once, never re-read -> NT stores
  float* vo = v_out + (size_t)pid * 3;
  float* xo = x_out + (size_t)pid * 3;
  __builtin_nontemporal_store(ax, vo + 0);
  __builtin_nontemporal_store(ay, vo + 1);
  __builtin_nontemporal_store(az, vo + 2);
  __builtin_nontemporal_store(fmaf(DT_STEP, ax, x0), xo + 0);
  __builtin_nontemporal_store(fmaf(DT_STEP, ay, x1), xo + 1);
  __builtin_nontemporal_store(fmaf(DT_STEP, az, x2), xo + 2);
}

extern "C" void kernel_launch(void* const* d_in, const int* in_sizes, int n_in,
                              void* d_out, int out_size, void* d_ws, size_t ws_size,
                              hipStream_t stream) {
  // inputs: x, v(unused), friction, pred, collider_floor, statics_enabled, step(unused)
  const float* x        = (const float*)d_in[0];
  const float* friction = (const float*)d_in[2];
  const float* pred     = (const float*)d_in[3];
  const float* cfloor   = (const float*)d_in[4];
  const int*   statics  = (const int*)d_in[5];

  int Bn = in_sizes[2];
  int Np = in_sizes[0] / (Bn * 3);
  long long gtot = (long long)in_sizes[3] / ((long long)Bn * 3);
  int G = 1;
  while ((long long)(G + 1) * (G + 1) * (G + 1) <= gtot) ++G;
  float dx = 1.0f / (float)G;

  float* x_out = (float*)d_out;
  float* v_out = x_out + (size_t)in_sizes[0];

  int totalP  = Bn * Np;
  int pBlocks = (totalP + BLOCKT - 1) / BLOCKT;

  size_t gridBytes = (size_t)Bn * G * G * G * 3 * sizeof(float);
  bool   pow2G     = (G & (G - 1)) == 0;

  if (ws_size >= gridBytes) {
    int totalN = Bn * G * G * G;
    if (pow2G) {
      int gShift  = __builtin_ctz((unsigned)G);
      int nChunks = totalN / CHUNK_NODES;
      int wBlocks = (nChunks + WPB - 1) / WPB;
      int nBlocks = wBlocks < 1 ? 1 : (wBlocks > 1024 ? 1024 : wBlocks);
      grid_op_async_kernel<<<nBlocks, BLOCKT, 0, stream>>>(
          pred, friction, cfloor, (float*)d_ws, totalN, gShift, G - 1, dx);
    } else {
      int nBlocks = (totalN + BLOCKT - 1) / BLOCKT;
      grid_op_kernel<<<nBlocks, BLOCKT, 0, stream>>>(
          pred, friction, cfloor, (float*)d_ws, Bn, G, dx);
    }
    g2p_kernel<false><<<pBlocks, BLOCKT, 0, stream>>>(
        x, (const float*)d_ws, friction, cfloor, statics,
        x_out, v_out, Bn, Np, G, dx, (float)G);
  } else {
    // fused fallback: apply grid-op inline per gathered node (no workspace)
    g2p_kernel<true><<<pBlocks, BLOCKT, 0, stream>>>(
        x, pred, friction, cfloor, statics,
        x_out, v_out, Bn, Np, G, dx, (float)G);
  }
}